// PriceAwareEnhancements_54769422959168
// MI455X (gfx1250) — compile-verified
//
#include <hip/hip_runtime.h>
#include <math.h>

typedef __attribute__((ext_vector_type(2))) float v2f;
typedef __attribute__((ext_vector_type(8))) float v8f;

#define EMB    128
#define LSEQ   50
#define NPRICE 100
#define NCAT   1000
#define ROWP   132   // padded LDS row stride (dwords); 132 % 64 = 4 -> conflict-free

// ---------------------------------------------------------------------------
// Kernel 0: repack conv_w [128 o][128 i][3 tap] into WMMA-B fragment order:
// wpack[((tap*32+kc)*8 + nt)*32 + lane] = { W'[k'=2h+0], W'[k'=2h+1] }
// where o = nt*16 + lane%16, i0 = kc*4 + 2*(lane/16), W'[(tap,i),o]=conv_w[o,i,tap]
// ---------------------------------------------------------------------------
__global__ void pack_weights(const float* __restrict__ conv_w,
                             float2* __restrict__ wpack) {
    int f = blockIdx.x * blockDim.x + threadIdx.x;
    if (f >= 3 * 32 * 8 * 32) return;
    int lane = f & 31;
    int nt   = (f >> 5) & 7;
    int kc   = (f >> 8) & 31;
    int tap  = f >> 13;
    int o  = nt * 16 + (lane & 15);
    int i0 = kc * 4 + 2 * (lane >> 4);
    float2 r;
    r.x = conv_w[o * 384 + (i0    ) * 3 + tap];
    r.y = conv_w[o * 384 + (i0 + 1) * 3 + tap];
    wpack[f] = r;
}

// ---------------------------------------------------------------------------
// Kernel 1: sensitivity gate (tiny: 384x128 matvec). One block, 128 threads.
// ---------------------------------------------------------------------------
__global__ void sens_kernel(const float* __restrict__ pe,
                            const float* __restrict__ ce,
                            const float* __restrict__ gate_w,
                            const float* __restrict__ gate_b,
                            float* __restrict__ out) {
    __shared__ float st[3 * EMB];
    int e = threadIdx.x;            // 0..127, one embedding dim each
    float s = 0.f, s2 = 0.f;
    for (int r = 0; r < NPRICE; ++r) {           // coalesced across threads
        float v = pe[r * EMB + e];
        s += v; s2 += v * v;
    }
    float mean = s * (1.0f / NPRICE);
    float var  = (s2 - (float)NPRICE * mean * mean) * (1.0f / (NPRICE - 1)); // ddof=1
    float sd   = sqrtf(var > 0.f ? var : 0.f);
    float cm = 0.f;
    for (int r = 0; r < NCAT; ++r) cm += ce[r * EMB + e];
    cm *= (1.0f / NCAT);
    st[e] = mean; st[EMB + e] = sd; st[2 * EMB + e] = cm;
    __syncthreads();
    float acc = gate_b[e];
    for (int k = 0; k < 3 * EMB; ++k) acc += st[k] * gate_w[e * 3 * EMB + k];
    out[e] = 1.0f / (1.0f + __expf(-acc));
}

// ---------------------------------------------------------------------------
// Kernel 2: per-sequence conv(k=3) as WMMA f32 16x16x4 GEMM + attention pool.
// One 256-thread block (8 waves) per batch row. Wave w owns N-tile w (16 out
// channels) x 4 M-tiles (time). K = 3 taps x 32 chunks of 4 input channels.
// ---------------------------------------------------------------------------
__global__ __launch_bounds__(256) void trend_kernel(
    const float*  __restrict__ pe,       // [100,128]
    const int*    __restrict__ seqs,     // [B,50]
    const int*    __restrict__ mask,     // [B,50]
    const float*  __restrict__ conv_b,   // [128]
    const float*  __restrict__ attn_w,   // [128]
    const float*  __restrict__ attn_b,   // [1]
    const float2* __restrict__ wpack,    // packed B fragments
    float*        __restrict__ out_trend)// [B,128]
{
    __shared__ float x_lds[66 * ROWP];   // rows: t=-1..64 (zero-padded), 34.8KB
    __shared__ float conv_lds[LSEQ * ROWP];               // 26.4KB
    __shared__ float aw[EMB];
    __shared__ float sw[64];             // scores -> softmax weights
    __shared__ int   vl_sh;

    const int tid  = threadIdx.x;
    const int b    = blockIdx.x;
    const int lane = tid & 31;
    const int w    = tid >> 5;           // wave id == N-tile

    if (tid == 0) vl_sh = 0;
    if (tid < EMB) aw[tid] = attn_w[tid];
    __syncthreads();
    if (tid < LSEQ) atomicAdd(&vl_sh, mask[b * LSEQ + tid] != 0 ? 1 : 0);

    // ---- gather embeddings into LDS (rows 1..50 valid, rest zero-pad) ----
    {
        int c = tid & 127;
        for (int r = (tid >> 7); r < 66; r += 2) {
            float v = 0.0f;
            if (r >= 1 && r <= LSEQ) {
                int s = seqs[b * LSEQ + (r - 1)];
                s = s < 0 ? 0 : (s > NPRICE - 1 ? NPRICE - 1 : s);
                v = pe[s * EMB + c];
            }
            x_lds[r * ROWP + c] = v;
        }
    }
    __syncthreads();

    // ---- conv as GEMM on the f32 WMMA pipe ----
    v8f acc[4];
    acc[0] = (v8f){}; acc[1] = (v8f){}; acc[2] = (v8f){}; acc[3] = (v8f){};
    const int col0  = 2 * (lane >> 4);   // K-pair select within chunk
    const int rlane = lane & 15;         // M within tile / N within tile

    for (int tap = 0; tap < 3; ++tap) {
        for (int kc = 0; kc < 32; ++kc) {
            v2f bf = *(const v2f*)&wpack[(((tap * 32 + kc) * 8) + w) * 32 + lane];
            int col   = kc * 4 + col0;
            int rbase = rlane + tap;
            v2f a0 = *(const v2f*)&x_lds[(rbase +  0) * ROWP + col];
            v2f a1 = *(const v2f*)&x_lds[(rbase + 16) * ROWP + col];
            v2f a2 = *(const v2f*)&x_lds[(rbase + 32) * ROWP + col];
            v2f a3 = *(const v2f*)&x_lds[(rbase + 48) * ROWP + col];
            acc[0] = __builtin_amdgcn_wmma_f32_16x16x4_f32(false, a0, false, bf, (short)0, acc[0], false, false);
            acc[1] = __builtin_amdgcn_wmma_f32_16x16x4_f32(false, a1, false, bf, (short)0, acc[1], false, false);
            acc[2] = __builtin_amdgcn_wmma_f32_16x16x4_f32(false, a2, false, bf, (short)0, acc[2], false, false);
            acc[3] = __builtin_amdgcn_wmma_f32_16x16x4_f32(false, a3, false, bf, (short)0, acc[3], false, false);
        }
    }

    // ---- bias + relu + store D fragments (mask t<50) ----
    {
        const int ocol  = w * 16 + rlane;
        const float cb  = conv_b[ocol];
        const int thalf = 8 * (lane >> 4);
        #pragma unroll
        for (int m = 0; m < 4; ++m) {
            #pragma unroll
            for (int v = 0; v < 8; ++v) {
                int t = m * 16 + v + thalf;
                if (t < LSEQ) {
                    float val = acc[m][v] + cb;
                    conv_lds[t * ROWP + ocol] = val > 0.f ? val : 0.f;
                }
            }
        }
    }
    __syncthreads();

    // ---- attention scores ----
    if (tid < LSEQ) {
        float s = attn_b[0];
        for (int c = 0; c < EMB; ++c) s += conv_lds[tid * ROWP + c] * aw[c];
        sw[tid] = s;
    }
    __syncthreads();
    if (tid == 0) {  // 50-wide softmax: serial is cheap
        float mx = sw[0];
        for (int t = 1; t < LSEQ; ++t) mx = fmaxf(mx, sw[t]);
        float sum = 0.f;
        for (int t = 0; t < LSEQ; ++t) { float e = __expf(sw[t] - mx); sw[t] = e; sum += e; }
        float inv = 1.0f / sum;
        for (int t = 0; t < LSEQ; ++t) sw[t] *= inv;
    }
    __syncthreads();

    // ---- pooled outputs + branch select ----
    if (tid < EMB) {
        int c = tid;
        float ap = 0.f, pool = 0.f;
        for (int t = 0; t < LSEQ; ++t) {
            ap   += conv_lds[t * ROWP + c] * sw[t];
            pool += x_lds[(t + 1) * ROWP + c];
        }
        pool *= (1.0f / LSEQ);
        int vl = vl_sh;
        float tr = (vl < 3) ? pool : ap;
        if (vl == 0) tr = 0.f;
        out_trend[b * EMB + c] = tr;
    }
}

// ---------------------------------------------------------------------------
extern "C" void kernel_launch(void* const* d_in, const int* in_sizes, int n_in,
                              void* d_out, int out_size, void* d_ws, size_t ws_size,
                              hipStream_t stream) {
    const float* pe      = (const float*)d_in[0];
    const int*   seqs    = (const int*)  d_in[1];
    const float* ce      = (const float*)d_in[2];
    const int*   mask    = (const int*)  d_in[3];
    const float* gate_w  = (const float*)d_in[4];
    const float* gate_b  = (const float*)d_in[5];
    const float* conv_w  = (const float*)d_in[6];
    const float* conv_b  = (const float*)d_in[7];
    const float* attn_w  = (const float*)d_in[8];
    const float* attn_b  = (const float*)d_in[9];
    float*  out   = (float*)d_out;
    float2* wpack = (float2*)d_ws;       // needs 3*32*8*32*8B = 192KB scratch

    const int B = in_sizes[1] / LSEQ;    // 4096

    pack_weights<<<(3 * 32 * 8 * 32 + 255) / 256, 256, 0, stream>>>(conv_w, wpack);
    sens_kernel<<<1, EMB, 0, stream>>>(pe, ce, gate_w, gate_b, out);
    trend_kernel<<<B, 256, 0, stream>>>(pe, seqs, mask, conv_b, attn_w, attn_b,
                                        wpack, out + EMB);
}